// WulfeniteLoss_71803263254994
// MI455X (gfx1250) — compile-verified
//
#include <hip/hip_runtime.h>

typedef float v2f __attribute__((ext_vector_type(2)));
typedef float v8f __attribute__((ext_vector_type(8)));

#define FS   160
#define NFR  1000
#define NB   64
#define EPSF 1e-8f

// ---------------------------------------------------------------------------
// Stage 1: frame energies via WMMA K-reduction.
// One wave handles 16 frames of one row of one signal.
// A (16x4 f32) = squared samples, B = ones  =>  D rows = frame sums.
// Two independent accumulator chains to break WMMA serialization.
// ---------------------------------------------------------------------------
__global__ void __launch_bounds__(128) frame_energy_wmma(
    const float* __restrict__ est, const float* __restrict__ tgt,
    const float* __restrict__ mix, float* __restrict__ ws)
{
    const int lane = threadIdx.x & 31;
    const int g    = blockIdx.x * 4 + (threadIdx.x >> 5);   // global wave id

    const int sig = g / (NB * 63);
    if (sig >= 3) return;                                    // uniform per wave
    const int rem = g - sig * (NB * 63);
    const int row = rem / 63;
    const int fg  = rem - row * 63;
    const int frameBase = fg * 16;

    const float* src = (sig == 0) ? est : (sig == 1) ? tgt : mix;

    // ISA A-matrix layout (32-bit, 16x4): M = lane%16; lanes<16 hold K={0,1},
    // lanes>=16 hold K={2,3}.
    const int M  = lane & 15;
    const int kh = lane >> 4;
    int frame  = frameBase + M;
    int frameC = frame < NFR ? frame : (NFR - 1);            // clamp; D rows independent
    const float* p = src + (size_t)row * (FS * NFR) + (size_t)frameC * FS + kh * 2;

    v8f acc0 = {};
    v8f acc1 = {};
    v2f ones = {1.0f, 1.0f};
#pragma unroll 8
    for (int kc = 0; kc < 40; kc += 2) {
        v2f x0 = *(const v2f*)(p + kc * 4);                  // 8B-aligned b64 loads
        v2f x1 = *(const v2f*)(p + kc * 4 + 4);
        v2f a0, a1;
        a0[0] = x0[0] * x0[0]; a0[1] = x0[1] * x0[1];
        a1[0] = x1[0] * x1[0]; a1[1] = x1[1] * x1[1];
        acc0 = __builtin_amdgcn_wmma_f32_16x16x4_f32(
            false, a0, false, ones, (short)0, acc0, false, false);
        acc1 = __builtin_amdgcn_wmma_f32_16x16x4_f32(
            false, a1, false, ones, (short)0, acc1, false, false);
    }
    v8f acc = acc0 + acc1;

    // D columns are identical (B = ones): every lane 0-15 holds E[fb+0..7] in
    // acc[0..7]; every lane 16-31 holds E[fb+8..15]. 16 lanes each pick one
    // element and issue a single coalesced predicated store.
    const int r = lane & 7;
    float v = (r & 4) ? ((r & 2) ? ((r & 1) ? acc[7] : acc[6])
                                 : ((r & 1) ? acc[5] : acc[4]))
                      : ((r & 2) ? ((r & 1) ? acc[3] : acc[2])
                                 : ((r & 1) ? acc[1] : acc[0]));
    const int f = frameBase + (kh << 3) + r;                 // lane>=16 -> +8
    const bool active = ((lane & 15) < 8) && (f < NFR);
    if (active) {
        float* dst = ws + (size_t)sig * (NB * NFR) + (size_t)row * NFR;
        dst[f] = v * (1.0f / FS);
    }
}

// ---------------------------------------------------------------------------
// Stage 2: per-scene masked means + global aggregation. One block.
// ---------------------------------------------------------------------------
__device__ __forceinline__ float wred32(float v) {
#pragma unroll
    for (int o = 16; o; o >>= 1) v += __shfl_xor(v, o, 32);
    return v;
}

__global__ void __launch_bounds__(1024) wulfenite_reduce(
    const float* __restrict__ ws,
    const int* __restrict__ ta,  const int* __restrict__ nta,
    const int* __restrict__ ov,  const int* __restrict__ bg,
    const int* __restrict__ scene_id, const int* __restrict__ role,
    float* __restrict__ out, int out_size)
{
    // 0:route_num 1:route_den 2:ovr_num 3:ovr_den 4:tt_num 5:tw_num 6:ab_den
    // 7:ot_num 8:ow_num 9:dom_den 10:n_pairs
    __shared__ float acc[11];
    if (threadIdx.x < 11) acc[threadIdx.x] = 0.0f;
    __syncthreads();

    const float* est_e = ws;
    const float* tgt_e = ws + NB * NFR;
    const float* mix_e = ws + 2 * NB * NFR;

    const int wave = threadIdx.x >> 5;
    const int lane = threadIdx.x & 31;

    for (int s = wave; s < NB; s += 32) {                    // scan all scene ids
        int aIdx = -1, bIdx = -1;
        for (int i = 0; i < NB; ++i) {
            if (scene_id[i] == s) {
                if (role[i] == 0 && aIdx < 0) aIdx = i;
                if (role[i] == 1 && bIdx < 0) bIdx = i;
            }
        }
        if (aIdx < 0 || bIdx < 0) continue;                  // invalid scene

        float s1=0,s2=0,s3=0,s4=0,s5=0,s6=0,s7=0,s8=0,s9=0,s10=0,s11=0,s12=0,s13=0;
        float cA=0,cB=0,cBG=0,cDA=0,cDB=0;

        for (int f = lane; f < NFR; f += 32) {
            float m  = mix_e[aIdx * NFR + f] + EPSF;
            float ar = est_e[aIdx * NFR + f] / m;
            float br = est_e[bIdx * NFR + f] / m;
            bool aonly = ta[aIdx * NFR + f] && !nta[aIdx * NFR + f];
            bool bonly = ta[bIdx * NFR + f] && !nta[bIdx * NFR + f];
            bool bgm   = bg[aIdx * NFR + f] && bg[bIdx * NFR + f];
            bool ovm   = ov[aIdx * NFR + f] && ov[bIdx * NFR + f];
            float gap  = (tgt_e[aIdx * NFR + f] - tgt_e[bIdx * NFR + f]) / m;
            bool da = ovm && (gap >  0.02f);
            bool db = ovm && (gap < -0.02f);
            float dab = ar - br;

            if (aonly) { float r = fmaxf(0.05f - dab, 0.0f); s1 += r*r; s6 += ar; s8 += br; cA += 1.0f; }
            if (bonly) { float r = fmaxf(0.05f + dab, 0.0f); s2 += r*r; s7 += br; s9 += ar; cB += 1.0f; }
            if (bgm)   { s3 += ar + br; cBG += 1.0f; }
            if (da)    { float r = fmaxf(0.02f - dab, 0.0f); s4 += r*r; s10 += ar; s12 += br; cDA += 1.0f; }
            if (db)    { float r = fmaxf(0.02f + dab, 0.0f); s5 += r*r; s11 += br; s13 += ar; cDB += 1.0f; }
        }

        s1 = wred32(s1);  s2 = wred32(s2);  s3 = wred32(s3);
        s4 = wred32(s4);  s5 = wred32(s5);  s6 = wred32(s6);
        s7 = wred32(s7);  s8 = wred32(s8);  s9 = wred32(s9);
        s10 = wred32(s10); s11 = wred32(s11); s12 = wred32(s12); s13 = wred32(s13);
        cA = wred32(cA);  cB = wred32(cB);  cBG = wred32(cBG);
        cDA = wred32(cDA); cDB = wred32(cDB);

        if (lane == 0) {
            bool fa = cA > 0, fb = cB > 0, fbg = cBG > 0, fda = cDA > 0, fdb = cDB > 0;
            atomicAdd(&acc[0], (fa ? s1/cA : 0.0f) + (fb ? s2/cB : 0.0f) + (fbg ? s3/cBG : 0.0f));
            atomicAdd(&acc[1], (float)((int)fa + (int)fb + (int)fbg));
            atomicAdd(&acc[2], (fda ? s4/cDA : 0.0f) + (fdb ? s5/cDB : 0.0f));
            atomicAdd(&acc[3], (float)((int)fda + (int)fdb));
            atomicAdd(&acc[4], (fa ? s6/cA : 0.0f) + (fb ? s7/cB : 0.0f));
            atomicAdd(&acc[5], (fa ? s8/cA : 0.0f) + (fb ? s9/cB : 0.0f));
            atomicAdd(&acc[6], (float)((int)fa + (int)fb));
            atomicAdd(&acc[7], (fda ? s10/cDA : 0.0f) + (fdb ? s11/cDB : 0.0f));
            atomicAdd(&acc[8], (fda ? s12/cDA : 0.0f) + (fdb ? s13/cDB : 0.0f));
            atomicAdd(&acc[9], (float)((int)fda + (int)fdb));
            atomicAdd(&acc[10], 1.0f);
        }
    }
    __syncthreads();

    if (threadIdx.x == 0) {
        float route = acc[1] > 0 ? acc[0] / acc[1] : 0.0f;
        float ovr   = acc[3] > 0 ? acc[2] / acc[3] : 0.0f;
        float tt    = acc[6] > 0 ? acc[4] / acc[6] : 0.0f;
        float tw    = acc[6] > 0 ? acc[5] / acc[6] : 0.0f;
        float ot    = acc[9] > 0 ? acc[7] / acc[9] : 0.0f;
        float ow    = acc[9] > 0 ? acc[8] / acc[9] : 0.0f;
        out[0] = route; out[1] = ovr;
        out[2] = tt;    out[3] = tw;  out[4] = tw;
        out[5] = ot;    out[6] = ow;
        out[7] = tt - tw; out[8] = ot - ow; out[9] = tw;
        if (out_size >= 11) out[10] = acc[10];               // n_pairs
    }
}

extern "C" void kernel_launch(void* const* d_in, const int* in_sizes, int n_in,
                              void* d_out, int out_size, void* d_ws, size_t ws_size,
                              hipStream_t stream) {
    const float* est = (const float*)d_in[0];
    const float* tgt = (const float*)d_in[1];
    const float* mix = (const float*)d_in[2];
    const int*   ta  = (const int*)d_in[3];
    const int*   nta = (const int*)d_in[4];
    const int*   ov  = (const int*)d_in[5];
    const int*   bg  = (const int*)d_in[6];
    const int*   sc  = (const int*)d_in[7];
    const int*   rl  = (const int*)d_in[8];
    float* ws  = (float*)d_ws;    // 3 * 64 * 1000 floats = 768 KB
    float* out = (float*)d_out;

    (void)in_sizes; (void)n_in; (void)ws_size;

    const int waves  = 3 * NB * 63;          // 12096 waves, 16 frames each
    const int blocks = waves / 4;            // 128 threads = 4 waves per block
    frame_energy_wmma<<<blocks, 128, 0, stream>>>(est, tgt, mix, ws);
    wulfenite_reduce<<<1, 1024, 0, stream>>>(ws, ta, nta, ov, bg, sc, rl, out, out_size);
}